// RGCN_78030965833745
// MI455X (gfx1250) — compile-verified
//
#include <hip/hip_runtime.h>
#include <hip/hip_bf16.h>

// Problem constants (match reference)
#define NN   100000   // num entities
#define DD   128      // embedding dim
#define HH   128      // hidden dim
#define RR2  100      // num relations * 2
#define BB   4        // num bases
#define EE   400000   // num edges
#define COLS 640      // BB*HH + HH  (concat of 4 bases + root)

typedef __attribute__((ext_vector_type(2))) float v2f;
typedef __attribute__((ext_vector_type(8))) float v8f;

// ---------------------------------------------------------------- utilities
__global__ void zero_u32_kernel(unsigned* __restrict__ p, int n) {
    int i = blockIdx.x * blockDim.x + threadIdx.x;
    if (i < n) p[i] = 0u;
}

// histogram of (dst, relation) pairs for schlichtkrull normalization
__global__ void count_kernel(const int* __restrict__ dst, const int* __restrict__ et,
                             unsigned* __restrict__ counts) {
    int e = blockIdx.x * blockDim.x + threadIdx.x;
    if (e >= EE) return;
    atomicAdd(&counts[(size_t)dst[e] * RR2 + et[e]], 1u);
}

__global__ void norm_kernel(const int* __restrict__ dst, const int* __restrict__ et,
                            const unsigned* __restrict__ counts, float* __restrict__ norm) {
    int e = blockIdx.x * blockDim.x + threadIdx.x;
    if (e >= EE) return;
    unsigned c = counts[(size_t)dst[e] * RR2 + et[e]];
    norm[e] = 1.0f / fmaxf((float)c, 1.0f);
}

// x0 = relu(emb + ebias)
__global__ void relu_bias_kernel(const float* __restrict__ emb, const float* __restrict__ ebias,
                                 float* __restrict__ x0) {
    int i = blockIdx.x * blockDim.x + threadIdx.x;
    if (i < NN * DD) x0[i] = fmaxf(emb[i] + ebias[i & (DD - 1)], 0.0f);
}

// Wcat[k][n] : n<512 -> bases[b=n>>7][k][n&127], else root[k][n-512]
__global__ void build_wcat_kernel(const float* __restrict__ bases, const float* __restrict__ root,
                                  float* __restrict__ Wcat) {
    int i = blockIdx.x * blockDim.x + threadIdx.x;
    if (i >= DD * COLS) return;
    int k = i / COLS, n = i % COLS;
    float v;
    if (n < BB * HH) v = bases[((n >> 7) * DD + k) * HH + (n & 127)];
    else             v = root[k * HH + (n - BB * HH)];
    Wcat[i] = v;
}

// h[n][j] = Y[n][512+j] + bias[j]   (root contribution + bias, before edge scatter)
__global__ void init_h_kernel(const float* __restrict__ Y, const float* __restrict__ bias,
                              float* __restrict__ h) {
    int i = blockIdx.x * blockDim.x + threadIdx.x;
    if (i >= NN * HH) return;
    int n = i >> 7, j = i & 127;
    h[i] = Y[(size_t)n * COLS + BB * HH + j] + bias[j];
}

__global__ void relu_inplace_kernel(float* __restrict__ p, int n) {
    int i = blockIdx.x * blockDim.x + threadIdx.x;
    if (i < n) p[i] = fmaxf(p[i], 0.0f);
}

// ---------------------------------------------------------------- WMMA GEMM
// Y[nrows x 640] = X[nrows x 128] @ W[128 x 640], fp32 via V_WMMA_F32_16X16X4_F32.
// Block = 8 waves = 8 row-tiles (128 rows) x one 80-col group. The W strip
// (128 x 80 = 40 KB) is staged once per block into LDS with async global->LDS
// copies (ASYNCcnt), then every wave feeds its B fragments from LDS.
// nrows % 16 == 0 (100000 = 6250*16) so EXEC is all-ones around every WMMA.
#define RT_PER_BLK 8     // row tiles (waves) per block
#define NT         5     // 16-col tiles per wave
#define CG         80    // columns per group (NT*16)

__global__ void __launch_bounds__(256) wmma_gemm_kernel(const float* __restrict__ X,
                                                        const float* __restrict__ W,
                                                        float* __restrict__ Y, int nrows) {
    __shared__ float lw[DD * CG];          // 40 KB: W[k][n0..n0+79], row-major
    const int lane = threadIdx.x & 31;
    const int wave = threadIdx.x >> 5;
    const int n0   = blockIdx.y * CG;

    // cooperative async stage of the W strip into LDS (16B chunks, contiguous
    // on both sides). 2560 chunks / 256 threads = 10 issues per thread.
    {
        const unsigned lbase = (unsigned)(uintptr_t)lw;
        for (int i = threadIdx.x; i < DD * (CG / 4); i += 256) {
            int k  = i / (CG / 4);
            int c4 = i - k * (CG / 4);
            unsigned long long ga =
                (unsigned long long)(uintptr_t)(W + (size_t)k * COLS + n0 + c4 * 4);
            unsigned la = lbase + (unsigned)i * 16u;
            asm volatile("global_load_async_to_lds_b128 %0, %1, off"
                         :: "v"(la), "v"(ga) : "memory");
        }
        asm volatile("s_wait_asynccnt 0x0" ::: "memory");
    }
    __syncthreads();

    const int tileRow = blockIdx.x * RT_PER_BLK + wave;
    if (tileRow >= (nrows >> 4)) return;   // wave-uniform, after the barrier

    const int row0 = tileRow << 4;
    const int mr   = lane & 15;            // A row / B col within tile
    const int kq   = (lane >> 4) << 1;     // lanes 0-15: k+{0,1}; 16-31: k+{2,3}

    const float* xr = X + (size_t)(row0 + mr) * DD + kq;
    const float* lb = lw + mr;

    v8f acc[NT] = {};
    for (int k = 0; k < DD; k += 4) {
        v2f a;
        a.x = xr[k];
        a.y = xr[k + 1];
        const float* lk = lb + (k + kq) * CG;
#pragma unroll
        for (int t = 0; t < NT; ++t) {
            v2f b;
            b.x = lk[t * 16];          // row k+kq   -> ds_load_2addr candidates
            b.y = lk[t * 16 + CG];     // row k+kq+1 (80 dwords apart)
            acc[t] = __builtin_amdgcn_wmma_f32_16x16x4_f32(
                false, a, false, b, (short)0, acc[t], false, false);
        }
    }
    // C/D layout: VGPR v, lanes 0-15 -> (M=v, N=lane); lanes 16-31 -> (M=v+8, N=lane-16)
    const int rBase = row0 + ((lane >> 4) << 3);
    const int cBase = n0 + mr;
#pragma unroll
    for (int t = 0; t < NT; ++t) {
        float* cb = Y + (size_t)rBase * COLS + cBase + t * 16;
#pragma unroll
        for (int v = 0; v < 8; ++v) cb[(size_t)v * COLS] = acc[t][v];
    }
}

// ---------------------------------------------------------------- edge scatter
// wave per edge: msg = (sum_b comp[et][b] * Y_b[src]) * norm, atomic-add into h[dst]
__global__ void __launch_bounds__(256) edge_msg_kernel(const float* __restrict__ Y,
                                                       const int* __restrict__ src,
                                                       const int* __restrict__ dst,
                                                       const int* __restrict__ et,
                                                       const float* __restrict__ comp,
                                                       const float* __restrict__ norm,
                                                       float* __restrict__ h) {
    int lane = threadIdx.x & 31;
    int e = (blockIdx.x << 3) + (threadIdx.x >> 5);
    e = __builtin_amdgcn_readfirstlane(e);   // provably wave-uniform -> scalar loads
    if (e >= EE) return;
    int s = src[e], d = dst[e], r = et[e];
    float c0 = comp[r * BB + 0], c1 = comp[r * BB + 1];
    float c2 = comp[r * BB + 2], c3 = comp[r * BB + 3];
    float nr = norm[e];
    const float* yb = Y + (size_t)s * COLS + (lane << 2);
    float4 y0 = *(const float4*)(yb);
    float4 y1 = *(const float4*)(yb + HH);
    float4 y2 = *(const float4*)(yb + 2 * HH);
    float4 y3 = *(const float4*)(yb + 3 * HH);
    float4 m;
    m.x = (c0 * y0.x + c1 * y1.x + c2 * y2.x + c3 * y3.x) * nr;
    m.y = (c0 * y0.y + c1 * y1.y + c2 * y2.y + c3 * y3.y) * nr;
    m.z = (c0 * y0.z + c1 * y1.z + c2 * y2.z + c3 * y3.z) * nr;
    m.w = (c0 * y0.w + c1 * y1.w + c2 * y2.w + c3 * y3.w) * nr;
    float* hd = h + (size_t)d * HH + (lane << 2);
    atomicAdd(hd + 0, m.x);
    atomicAdd(hd + 1, m.y);
    atomicAdd(hd + 2, m.z);
    atomicAdd(hd + 3, m.w);
}

// ---------------------------------------------------------------- decoder
__global__ void __launch_bounds__(256) decoder_kernel(const float* __restrict__ x,
                                                      const int* __restrict__ src,
                                                      const int* __restrict__ dst,
                                                      const int* __restrict__ et,
                                                      const float* __restrict__ rel,
                                                      float* __restrict__ out) {
    int lane = threadIdx.x & 31;
    int e = (blockIdx.x << 3) + (threadIdx.x >> 5);
    e = __builtin_amdgcn_readfirstlane(e);
    if (e >= EE) return;
    const float4* xs = (const float4*)(x + (size_t)src[e] * DD);
    const float4* xd = (const float4*)(x + (size_t)dst[e] * DD);
    const float4* rr = (const float4*)(rel + (size_t)et[e] * DD);
    float4 a = xs[lane], b = xd[lane], c = rr[lane];
    float p = a.x * b.x * c.x + a.y * b.y * c.y + a.z * b.z * c.z + a.w * b.w * c.w;
#pragma unroll
    for (int m = 16; m; m >>= 1) p += __shfl_xor(p, m, 32);
    if (lane == 0) out[e] = p;
}

__global__ void penalty_kernel(const float* __restrict__ rel, float* __restrict__ out) {
    __shared__ float red[256];
    float s = 0.0f;
    for (int i = threadIdx.x; i < RR2 * DD; i += 256) { float v = rel[i]; s += v * v; }
    red[threadIdx.x] = s;
    __syncthreads();
    for (int k = 128; k; k >>= 1) {
        if ((int)threadIdx.x < k) red[threadIdx.x] += red[threadIdx.x + k];
        __syncthreads();
    }
    if (threadIdx.x == 0) out[EE] = red[0];
}

// ---------------------------------------------------------------- launch
extern "C" void kernel_launch(void* const* d_in, const int* in_sizes, int n_in,
                              void* d_out, int out_size, void* d_ws, size_t ws_size,
                              hipStream_t stream) {
    const int*   edge_index = (const int*)d_in[0];       // [2, E]
    const int*   src   = edge_index;
    const int*   dst   = edge_index + EE;
    const int*   et    = (const int*)d_in[1];            // [E]
    const float* emb   = (const float*)d_in[2];          // [N, D]
    const float* ebias = (const float*)d_in[3];          // [1, D]
    const float* bases1= (const float*)d_in[4];          // [B, D, H]
    const float* comp1 = (const float*)d_in[5];          // [R2, B]
    const float* root1 = (const float*)d_in[6];          // [D, H]
    const float* bias1 = (const float*)d_in[7];          // [H]
    const float* bases2= (const float*)d_in[8];          // [B, H, D]
    const float* comp2 = (const float*)d_in[9];          // [R2, B]
    const float* root2 = (const float*)d_in[10];         // [H, D]
    const float* bias2 = (const float*)d_in[11];         // [D]
    const float* rel   = (const float*)d_in[12];         // [R2, D]
    float* out = (float*)d_out;                          // [E] scores + [1] penalty

    // workspace carve-out (~400 MB)
    char* w = (char*)d_ws;
    size_t off = 0;
    auto carve = [&](size_t bytes) -> char* {
        char* p = w + off;
        off += (bytes + 255) & ~(size_t)255;
        return p;
    };
    unsigned* counts = (unsigned*)carve((size_t)NN * RR2 * 4);   // 40 MB
    float*    norm   = (float*)carve((size_t)EE * 4);            // 1.6 MB
    float*    xbuf   = (float*)carve((size_t)NN * DD * 4);       // 51.2 MB (x0, reused as x2)
    float*    h1     = (float*)carve((size_t)NN * HH * 4);       // 51.2 MB
    float*    Ybuf   = (float*)carve((size_t)NN * COLS * 4);     // 256 MB
    float*    Wcat   = (float*)carve((size_t)DD * COLS * 4);     // 0.33 MB
    (void)ws_size; (void)n_in; (void)in_sizes; (void)out_size;

    const int T = 256;
    const int nc = NN * RR2;

    // normalization
    zero_u32_kernel<<<(nc + T - 1) / T, T, 0, stream>>>(counts, nc);
    count_kernel<<<(EE + T - 1) / T, T, 0, stream>>>(dst, et, counts);
    norm_kernel<<<(EE + T - 1) / T, T, 0, stream>>>(dst, et, counts, norm);

    // x0 = relu(emb + bias)
    relu_bias_kernel<<<(NN * DD + T - 1) / T, T, 0, stream>>>(emb, ebias, xbuf);

    dim3 gemmGrid(((NN >> 4) + RT_PER_BLK - 1) / RT_PER_BLK, COLS / CG);   // (782, 8)
    const int edgeBlocks = EE / 8;                                         // 50000

    // ---- layer 1 ----
    build_wcat_kernel<<<(DD * COLS + T - 1) / T, T, 0, stream>>>(bases1, root1, Wcat);
    wmma_gemm_kernel<<<gemmGrid, T, 0, stream>>>(xbuf, Wcat, Ybuf, NN);
    init_h_kernel<<<(NN * HH + T - 1) / T, T, 0, stream>>>(Ybuf, bias1, h1);
    edge_msg_kernel<<<edgeBlocks, T, 0, stream>>>(Ybuf, src, dst, et, comp1, norm, h1);
    relu_inplace_kernel<<<(NN * HH + T - 1) / T, T, 0, stream>>>(h1, NN * HH);

    // ---- layer 2 ----  (xbuf reused as output x2)
    build_wcat_kernel<<<(DD * COLS + T - 1) / T, T, 0, stream>>>(bases2, root2, Wcat);
    wmma_gemm_kernel<<<gemmGrid, T, 0, stream>>>(h1, Wcat, Ybuf, NN);
    init_h_kernel<<<(NN * DD + T - 1) / T, T, 0, stream>>>(Ybuf, bias2, xbuf);
    edge_msg_kernel<<<edgeBlocks, T, 0, stream>>>(Ybuf, src, dst, et, comp2, norm, xbuf);

    // ---- DistMult decoder + penalty ----
    decoder_kernel<<<edgeBlocks, T, 0, stream>>>(xbuf, src, dst, et, rel, out);
    penalty_kernel<<<1, T, 0, stream>>>(rel, out);
}